// LocalKNNFeastBlock_66133906424006
// MI455X (gfx1250) — compile-verified
//
#include <hip/hip_runtime.h>
#include <math.h>

typedef float v2f __attribute__((ext_vector_type(2)));
typedef float v8f __attribute__((ext_vector_type(8)));

#define N_NODES 50000
#define DIM     128
#define NHEAD   8
#define DHEAD   16
#define KNN     32
#define FFND    192
#define ROWS    80          // 5 row-tiles of 16; 50000 = 625 * 80
#define RT      5           // row tiles per block

// ---------------------------------------------------------------------------
// fp32 WMMA fragment layout (CDNA5, wave32):
//   A(16x4): lane(0-15)=row r, kh=lane>>4; VGPR0=K(2kh), VGPR1=K(2kh+1)
//   B(4x16): lane(0-15)=col c, same K striping
//   C/D:     VGPR v -> row v+8*kh, col = lane&15
// Packed weights: P[(ct*(K/4)+ks)*32+lane] = {W[(4ks+2kh)*NCOL+c], W[..+1..]}
// -> one coalesced global_load_b64 per wave per k-step.
// ---------------------------------------------------------------------------

__global__ __launch_bounds__(256)
void repack_kernel(const float* __restrict__ W, float2* __restrict__ P,
                   int KDIM, int NCOL)
{
    const int total = (KDIM / 4) * 32 * (NCOL / 16);
    for (int p = blockIdx.x * blockDim.x + threadIdx.x; p < total;
         p += gridDim.x * blockDim.x) {
        const int lane = p & 31;
        const int ks   = (p >> 5) % (KDIM / 4);
        const int ct   = (p >> 5) / (KDIM / 4);
        const int c    = ct * 16 + (lane & 15);
        const int row  = 4 * ks + 2 * (lane >> 4);
        P[p] = make_float2(W[(size_t)row * NCOL + c],
                           W[(size_t)(row + 1) * NCOL + c]);
    }
}

// MMA core: 5 row-tiles x 1 col-tile, A from LDS, packed B from global.
// Unroll capped at 8 k-steps: 2KB of B loads in flight per wave is enough
// pipelining while keeping VGPR pressure low enough for multi-wave residency.
template <int KDIM>
__device__ __forceinline__ void mma5(const float* __restrict__ As,
                                     const float2* __restrict__ Bt,
                                     v8f acc[RT], int r, int kh, int lane)
{
    #pragma unroll 8
    for (int ks = 0; ks < KDIM / 4; ++ks) {
        if ((ks & 7) == 0)   // uniform branch: EXEC stays all-ones for WMMA
            __builtin_prefetch((const void*)(Bt + ks * 32 + 256), 0, 3);
        const float2 bb = Bt[ks * 32 + lane];
        v2f b; b.x = bb.x; b.y = bb.y;
        const int kk = 4 * ks + 2 * kh;
        #pragma unroll
        for (int t = 0; t < RT; ++t) {
            v2f a;
            a.x = As[(t * 16 + r) * KDIM + kk];
            a.y = As[(t * 16 + r) * KDIM + kk + 1];
            acc[t] = __builtin_amdgcn_wmma_f32_16x16x4_f32(false, a, false, b,
                                                           (short)0, acc[t],
                                                           false, false);
        }
    }
}

// ============================ Kernel 1: QKV ================================
__global__ __launch_bounds__(256)
void qkv_kernel(const float* __restrict__ feat, const float2* __restrict__ Pqkv,
                float* __restrict__ Qo, float* __restrict__ Ko, float* __restrict__ Vo)
{
    __shared__ float As[ROWS * DIM];
    const int row0 = blockIdx.x * ROWS;
    const int tid  = threadIdx.x;

    for (int i = tid; i < ROWS * DIM / 4; i += 256)
        ((float4*)As)[i] = ((const float4*)(feat + (size_t)row0 * DIM))[i];
    __syncthreads();

    const int wave = tid >> 5, lane = tid & 31;
    const int r = lane & 15, kh = lane >> 4;

    for (int t = wave; t < 24; t += 8) {               // 24 col tiles: Q|K|V
        float* Out = (t < 8) ? Qo : (t < 16) ? Ko : Vo;
        const int ct = t & 7;
        const float2* Bt = Pqkv + (size_t)t * (DIM / 4) * 32;
        v8f acc[RT] = {};
        mma5<DIM>(As, Bt, acc, r, kh, lane);
        const int c = ct * 16 + r;
        #pragma unroll
        for (int t5 = 0; t5 < RT; ++t5)
            #pragma unroll
            for (int v = 0; v < 8; ++v)
                Out[(size_t)(row0 + t5 * 16 + v + 8 * kh) * DIM + c] = acc[t5][v];
    }
}

// ========================= Kernel 2: attention =============================
// One wave per node; lane == neighbor index (K=32 == wave32). L2-gather bound.
__global__ __launch_bounds__(256)
void attn_kernel(const float* __restrict__ Q, const float* __restrict__ Kp,
                 const float* __restrict__ Vp, const int* __restrict__ knn,
                 const float* __restrict__ dist, float* __restrict__ att)
{
    const int tid  = threadIdx.x;
    const int wave = tid >> 5, lane = tid & 31;
    const int n    = blockIdx.x * 8 + wave;
    if (n >= N_NODES) return;

    const int   idx = knn[(size_t)n * KNN + lane];
    const float dd  = dist[(size_t)n * KNN + lane];
    const float* krow = Kp + (size_t)idx * DIM;
    const float* vrow = Vp + (size_t)idx * DIM;
    const float* qrow = Q  + (size_t)n   * DIM;

    float scale = 0.5f;                       // 0.5^(h+1)
    for (int h = 0; h < NHEAD; ++h) {
        const float* qh = qrow + h * DHEAD;   // wave-uniform -> scalar loads
        const float* kh = krow + h * DHEAD;
        float s = 0.f;
        #pragma unroll
        for (int d = 0; d < DHEAD; ++d) s = fmaf(qh[d], kh[d], s);
        s *= 0.25f;

        float sp = s - dd * scale;
        float sn = (-s - dd * scale) * (1.0f / 0.6f);

        float mp = sp, mn = sn;
        #pragma unroll
        for (int off = 16; off; off >>= 1) {
            mp = fmaxf(mp, __shfl_xor(mp, off, 32));
            mn = fmaxf(mn, __shfl_xor(mn, off, 32));
        }
        float ep = expf(sp - mp), en = expf(sn - mn);
        float sump = ep, sumn = en;
        #pragma unroll
        for (int off = 16; off; off >>= 1) {
            sump += __shfl_xor(sump, off, 32);
            sumn += __shfl_xor(sumn, off, 32);
        }
        const float a = ep / sump - 1.5f * en / sumn;

        const float* vh = vrow + h * DHEAD;
        float accv[DHEAD];
        #pragma unroll
        for (int d = 0; d < DHEAD; ++d) accv[d] = a * vh[d];
        #pragma unroll
        for (int off = 16; off; off >>= 1)
            #pragma unroll
            for (int d = 0; d < DHEAD; ++d)
                accv[d] += __shfl_xor(accv[d], off, 32);

        if (lane == 0) {
            #pragma unroll
            for (int d = 0; d < DHEAD; ++d)
                att[(size_t)n * DIM + h * DHEAD + d] = accv[d];
        }
        scale *= 0.5f;
    }
}

// ============== Kernels 3/5: GEMM (KDIM->128) + bias + residual + LN =======
template <int KDIM>
__global__ __launch_bounds__(256)
void gemm_ln_kernel(const float* __restrict__ Ag, const float2* __restrict__ Pw,
                    const float* __restrict__ bias, const float* __restrict__ resid,
                    const float* __restrict__ lng, const float* __restrict__ lnb,
                    float* __restrict__ outp)
{
    __shared__ float S[ROWS * KDIM];      // A-stage, reused as D-stage (KDIM>=DIM)
    const int row0 = blockIdx.x * ROWS;
    const int tid  = threadIdx.x;

    for (int i = tid; i < ROWS * KDIM / 4; i += 256)
        ((float4*)S)[i] = ((const float4*)(Ag + (size_t)row0 * KDIM))[i];
    __syncthreads();

    const int wave = tid >> 5, lane = tid & 31;
    const int r = lane & 15, kh = lane >> 4;
    const int c = wave * 16 + r;                        // 8 waves -> 128 cols

    v8f acc[RT] = {};
    mma5<KDIM>(S, Pw + (size_t)wave * (KDIM / 4) * 32, acc, r, kh, lane);
    __syncthreads();                                    // all A reads done
    #pragma unroll
    for (int t5 = 0; t5 < RT; ++t5)
        #pragma unroll
        for (int v = 0; v < 8; ++v)
            S[(t5 * 16 + v + 8 * kh) * DIM + c] = acc[t5][v];
    __syncthreads();

    // LayerNorm: 16 threads/row (width-16 shuffles), 5 groups of 16 rows.
    const int seg = tid & 15;
    for (int it = 0; it < RT; ++it) {
        const int rrow = it * 16 + (tid >> 4);
        float vals[8], sum = 0.f, sumsq = 0.f;
        #pragma unroll
        for (int j = 0; j < 8; ++j) {
            const int d = seg * 8 + j;
            const float v = S[rrow * DIM + d] + bias[d]
                          + resid[(size_t)(row0 + rrow) * DIM + d];
            vals[j] = v; sum += v; sumsq = fmaf(v, v, sumsq);
        }
        #pragma unroll
        for (int off = 8; off; off >>= 1) {
            sum   += __shfl_xor(sum,   off, 16);
            sumsq += __shfl_xor(sumsq, off, 16);
        }
        const float mean = sum * (1.0f / DIM);
        const float var  = sumsq * (1.0f / DIM) - mean * mean;
        const float rstd = rsqrtf(var + 1e-5f);
        #pragma unroll
        for (int j = 0; j < 8; ++j) {
            const int d = seg * 8 + j;
            outp[(size_t)(row0 + rrow) * DIM + d] =
                (vals[j] - mean) * rstd * lng[d] + lnb[d];
        }
    }
}

// ===================== Kernel 4: FFN1 (128->192) + GELU ====================
__global__ __launch_bounds__(384)
void ffn1_kernel(const float* __restrict__ X, const float2* __restrict__ P1,
                 const float* __restrict__ b1, float* __restrict__ Hout)
{
    __shared__ float As[ROWS * DIM];
    const int row0 = blockIdx.x * ROWS;
    const int tid  = threadIdx.x;
    for (int i = tid; i < ROWS * DIM / 4; i += 384)
        ((float4*)As)[i] = ((const float4*)(X + (size_t)row0 * DIM))[i];
    __syncthreads();

    const int wave = tid >> 5, lane = tid & 31;        // 12 waves -> 192 cols
    const int r = lane & 15, kh = lane >> 4;
    const int c = wave * 16 + r;

    v8f acc[RT] = {};
    mma5<DIM>(As, P1 + (size_t)wave * (DIM / 4) * 32, acc, r, kh, lane);

    const float bb = b1[c];
    #pragma unroll
    for (int t5 = 0; t5 < RT; ++t5)
        #pragma unroll
        for (int v = 0; v < 8; ++v) {
            const float x  = acc[t5][v] + bb;
            const float ge = 0.5f * x * (1.0f + erff(x * 0.70710678118654752f));
            Hout[(size_t)(row0 + t5 * 16 + v + 8 * kh) * FFND + c] = ge;
        }
}

// ================================ Launch ===================================
extern "C" void kernel_launch(void* const* d_in, const int* in_sizes, int n_in,
                              void* d_out, int out_size, void* d_ws, size_t ws_size,
                              hipStream_t stream)
{
    const float* feat = (const float*)d_in[0];
    const int*   knn  = (const int*)  d_in[1];
    const float* dist = (const float*)d_in[2];
    const float* Wq   = (const float*)d_in[3];
    const float* Wk   = (const float*)d_in[4];
    const float* Wv   = (const float*)d_in[5];
    const float* Wo   = (const float*)d_in[6];
    const float* bo   = (const float*)d_in[7];
    const float* ln1g = (const float*)d_in[8];
    const float* ln1b = (const float*)d_in[9];
    const float* W1   = (const float*)d_in[10];
    const float* b1   = (const float*)d_in[11];
    const float* W2   = (const float*)d_in[12];
    const float* b2   = (const float*)d_in[13];
    const float* ln2g = (const float*)d_in[14];
    const float* ln2b = (const float*)d_in[15];

    float* ws = (float*)d_ws;
    const size_t SZ = (size_t)N_NODES * DIM;
    float* Q   = ws;                 // dead after attn -> reused for X
    float* Kp  = ws + SZ;            // dead after attn -> reused for H
    float* Vp  = ws + 2 * SZ;
    float* att = ws + 3 * SZ;
    float* X   = ws;                 // alias Q
    float* Hb  = ws + SZ;            // alias Kp..Vp (N*192 <= 2*N*128)

    // Packed weights (fragment order), after att region.
    float*  P    = ws + 4 * SZ;
    float2* Pqkv = (float2*)P;              // 24 tiles * 1024 f2 = 24576 f2
    float2* Po   = Pqkv + 24576;            // 8192 f2
    float2* P1   = Po   + 8192;             // 12288 f2
    float2* P2   = P1   + 12288;            // 12288 f2

    repack_kernel<<<32, 256, 0, stream>>>(Wq, Pqkv,          DIM,  DIM);
    repack_kernel<<<32, 256, 0, stream>>>(Wk, Pqkv +  8192,  DIM,  DIM);
    repack_kernel<<<32, 256, 0, stream>>>(Wv, Pqkv + 16384,  DIM,  DIM);
    repack_kernel<<<32, 256, 0, stream>>>(Wo, Po,            DIM,  DIM);
    repack_kernel<<<48, 256, 0, stream>>>(W1, P1,            DIM,  FFND);
    repack_kernel<<<48, 256, 0, stream>>>(W2, P2,            FFND, DIM);

    const int nbR = N_NODES / ROWS;  // 625
    const int nb8 = (N_NODES + 7) / 8;

    qkv_kernel<<<nbR, 256, 0, stream>>>(feat, Pqkv, Q, Kp, Vp);
    attn_kernel<<<nb8, 256, 0, stream>>>(Q, Kp, Vp, knn, dist, att);
    gemm_ln_kernel<DIM ><<<nbR, 256, 0, stream>>>(att, Po, bo, feat, ln1g, ln1b, X);
    ffn1_kernel<<<nbR, 384, 0, stream>>>(X, P1, b1, Hb);
    gemm_ln_kernel<FFND><<<nbR, 256, 0, stream>>>(Hb, P2, b2, X, ln2g, ln2b, (float*)d_out);
}